// ModelNew_22943715295862
// MI455X (gfx1250) — compile-verified
//
#include <hip/hip_runtime.h>
#include <stdint.h>

// masked row-wise cumsum: out[m, :] = cumsum(x[m,:] * mask[m,:]), fp32.
// M = 4096 rows, N = 32768 cols. Pure HBM-streaming problem (~1.15 GB total
// traffic -> ~50us floor at 23.3 TB/s). One workgroup (8 wave32) per row,
// sequential 1024-element chunks. Chunk staging uses the gfx1250 Tensor Data
// Mover (TENSOR_LOAD_TO_LDS, TENSORcnt-tracked) with double buffering: wave 0
// issues one DMA per chunk for x and one for mask while all waves scan the
// previous chunk out of LDS.

#define NCOLS   32768
#define BLOCK   256
#define VEC     4
#define CHUNK   (BLOCK * VEC)     // 1024 elements per chunk
#define NCHUNK  (NCOLS / CHUNK)   // 32 chunks per row
#define NWAVES  (BLOCK / 32)      // 8 wave32 per block

typedef float v4f __attribute__((ext_vector_type(4)));
typedef unsigned int u32x4 __attribute__((ext_vector_type(4)));
typedef int i32x4 __attribute__((ext_vector_type(4)));
typedef int i32x8 __attribute__((ext_vector_type(8)));

#if __has_builtin(__builtin_amdgcn_tensor_load_to_lds)
#define HAVE_TDM 1
#else
#define HAVE_TDM 0
#endif

// 32-bit LDS byte address of a __shared__ object (generic -> AS(3) -> int).
__device__ __forceinline__ unsigned lds_addr_of(const void* p) {
  return (unsigned)(size_t)(__attribute__((address_space(3))) const char*)p;
}

#if HAVE_TDM
// Issue a 1-D TDM transfer: n_elems elements of (1<<ds_code) bytes each,
// global -> LDS. Descriptor per CDNA5 ISA 8.3/8.4 (D# groups 0/1; groups 2/3
// NULL-equivalent zeros: tile_dim1==0 -> 1-D tile).
__device__ __forceinline__ void tdm_load_1d(unsigned lds_addr,
                                            unsigned long long gaddr,
                                            unsigned n_elems,
                                            unsigned ds_code) {
  u32x4 g0;
  g0.x = 0x1u;                                    // count=1, user descriptor
  g0.y = lds_addr;                                // lds_addr (bytes)
  g0.z = (unsigned)(gaddr & 0xFFFFFFFFull);       // global_addr[31:0]
  g0.w = (unsigned)((gaddr >> 32) & 0x01FFFFFFull)// global_addr[56:32]
         | (2u << 30);                            // type = 2 ("image")
  i32x8 g1;
  g1[0] = (int)(ds_code << 16);                   // wg_mask=0, data_size
  g1[1] = (int)((n_elems & 0xFFFFu) << 16);       // tensor_dim0[15:0]
  g1[2] = (int)((n_elems >> 16) & 0xFFFFu);       // tensor_dim0[31:16]
  g1[3] = (int)((n_elems & 0xFFFFu) << 16);       // tile_dim0
  g1[4] = 0;                                      // tile_dim1=0 (1-D), tile_dim2=0
  g1[5] = (int)n_elems;                           // dim0 stride (ignored, 1-D)
  g1[6] = 0;
  g1[7] = 0;
  i32x4 z4 = {0, 0, 0, 0};
#if defined(__clang_major__) && __clang_major__ >= 23
  i32x8 z8 = {0, 0, 0, 0, 0, 0, 0, 0};
  __builtin_amdgcn_tensor_load_to_lds(g0, g1, z4, z4, z8, 0);
#else
  __builtin_amdgcn_tensor_load_to_lds(g0, g1, z4, z4, 0);
#endif
}
#endif

__device__ __forceinline__ void wait_tensor0() {
#if __has_builtin(__builtin_amdgcn_s_wait_tensorcnt)
  __builtin_amdgcn_s_wait_tensorcnt(0);
#elif HAVE_TDM
  asm volatile("s_wait_tensorcnt 0x0" ::: "memory");
#endif
}

// --------------------------------------------------------------------------
__global__ __launch_bounds__(BLOCK)
void masked_cumsum_rows(const float* __restrict__ x,
                        const unsigned char* __restrict__ mask,
                        float* __restrict__ out, int rows) {
  __shared__ float         xbuf[2][CHUNK];   // 2 x 4 KiB
  __shared__ unsigned char mbuf[2][CHUNK];   // 2 x 1 KiB
  __shared__ float         wtot[NWAVES];

  const int row = blockIdx.x;
  if (row >= rows) return;                    // uniform across the block
  const size_t base = (size_t)row * (size_t)NCOLS;
  const int t    = threadIdx.x;
  const int lane = t & 31;
  const int wid  = t >> 5;

  // Stage a chunk into LDS buffer `b`.
  auto stage = [&](int c, int b) {
    const size_t off = base + (size_t)c * CHUNK;
#if HAVE_TDM
    if (wid == 0) {
      tdm_load_1d(lds_addr_of(&xbuf[b][0]),
                  (unsigned long long)(uintptr_t)(x + off), CHUNK, 2u);  // 4 B
      tdm_load_1d(lds_addr_of(&mbuf[b][0]),
                  (unsigned long long)(uintptr_t)(mask + off), CHUNK, 0u); // 1 B
    }
#else
    *(v4f*)&xbuf[b][t * VEC] = *(const v4f*)(x + off + (size_t)t * VEC);
    *(unsigned*)&mbuf[b][t * 4] = *(const unsigned*)(mask + off + (size_t)t * 4);
#endif
  };

  // Prologue: stage chunk 0 into buffer 0.
  stage(0, 0);
  wait_tensor0();
  __syncthreads();

  float carry = 0.0f;

  for (int c = 0; c < NCHUNK; ++c) {
    const int cur = c & 1;
    const int nxt = cur ^ 1;

    // Kick off the DMA of the next chunk into the other buffer.
    if (c + 1 < NCHUNK) stage(c + 1, nxt);

    // Per-thread 4-element masked values + serial prefix.
    v4f xv = *(const v4f*)&xbuf[cur][t * VEC];            // ds_load_b128
    unsigned mm = *(const unsigned*)&mbuf[cur][t * 4];    // 4 mask bytes
    const float v0 = xv.x * (float)( mm        & 1u);
    const float v1 = xv.y * (float)((mm >> 8)  & 1u);
    const float v2 = xv.z * (float)((mm >> 16) & 1u);
    const float v3 = xv.w * (float)((mm >> 24) & 1u);
    const float p0 = v0;
    const float p1 = p0 + v1;
    const float p2 = p1 + v2;
    const float p3 = p2 + v3;
    const float tsum = p3;

    // Wave32 inclusive scan of per-thread sums (Hillis-Steele via shuffles).
    float s = tsum;
#pragma unroll
    for (int d = 1; d < 32; d <<= 1) {
      float n = __shfl_up(s, d, 32);
      if (lane >= d) s += n;
    }
    if (lane == 31) wtot[wid] = s;
    __syncthreads();

    // Cross-wave offsets + chunk total (8 broadcast LDS reads, all lanes).
    float off   = carry;
    float total = 0.0f;
#pragma unroll
    for (int w = 0; w < NWAVES; ++w) {
      const float wv = wtot[w];
      total += wv;
      if (w < wid) off += wv;
    }

    const float excl = off + (s - tsum);  // exclusive prefix for this thread
    v4f o;
    o.x = excl + p0;
    o.y = excl + p1;
    o.z = excl + p2;
    o.w = excl + p3;
    const size_t oidx = base + (size_t)c * CHUNK + (size_t)t * VEC;
    __builtin_nontemporal_store(o, (v4f*)(out + oidx));   // stream-once: NT

    carry += total;

    // Issuing wave drains its TENSORcnt; barrier makes the new buffer
    // visible to all waves and retires all reads of buf[cur] before it is
    // overwritten next iteration.
    wait_tensor0();
    __syncthreads();
  }
}

// --------------------------------------------------------------------------
extern "C" void kernel_launch(void* const* d_in, const int* in_sizes, int n_in,
                              void* d_out, int out_size, void* d_ws, size_t ws_size,
                              hipStream_t stream) {
  const float*         x    = (const float*)d_in[0];
  const unsigned char* mask = (const unsigned char*)d_in[1];  // jnp.bool_: 1 byte/elem
  float*               out  = (float*)d_out;

  const int rows = out_size / NCOLS;  // 4096
  dim3 grid(rows), block(BLOCK);
  hipLaunchKernelGGL(masked_cumsum_rows, grid, block, 0, stream,
                     x, mask, out, rows);
}